// DHGCN_AdaptConv_53893249630262
// MI455X (gfx1250) — compile-verified
//
#include <hip/hip_runtime.h>
#include <hip/hip_bf16.h>

// ---------------------------------------------------------------------------
// DHGCN forward for MI455X (gfx1250, wave32).  Heavy matmuls run through
// V_WMMA_F32_16X16X4_F32 (full fp32 preserves top-k / argmax decisions).
// GEMM: per block, one 64-wide output strip; W staged zero-padded in LDS;
// each wave computes 16x64 via 4 accumulators sharing one A fragment.
// All hot-loop loads use clamped addresses + cndmask select (no exec-mask
// divergence in the K loop).
// ---------------------------------------------------------------------------

typedef __attribute__((ext_vector_type(2))) float v2f;
typedef __attribute__((ext_vector_type(8))) float v8f;

#define BN_INV 0.99999500003749973f   // 1/sqrt(1+1e-5)
#define GAUSS_C 0.3989422804014327f   // 1/sqrt(2*pi)
#define KC 128                        // K-chunk staged in LDS (64*KC*4 = 32KB)

__device__ __forceinline__ float lrelu_f(float x) { return x > 0.f ? x : 0.2f * x; }
__device__ __forceinline__ int imin_i(int a, int b) { return a < b ? a : b; }

// ---------------------------------------------------------------------------
// Generic fp32 WMMA GEMM:  Out[M,O] = act( bn( A[M,K] * W[O,K]^T ) )
// A-source modes:
//   0 = dense rows A[M,K]
//   1 = graph feature gather: row=(n*kk+q):
//         c<C  -> F[c*N + idx[row]] - F[c*N + n] ; else F[(c-C)*N + n]
//   2 = part-edge feature: row=(i*P+j):
//         c<C  -> F[c*P + j] - F[c*P + i]        ; else F[(c-C)*P + i]
// ---------------------------------------------------------------------------
struct GemmP {
  const float* A;
  const float* F;
  const int*   IDX;
  const float* W;     // [O,K] row-major
  const float* bg;    // bn gamma (or null)
  const float* bb;    // bn beta  (or null)
  float*       Out;   // [M,O]
  int M, K, O;
  int N, kk, C, P;
  int act;
};

template <int MODE>
__global__ void gemm_wmma_k(GemmP p) {
  extern __shared__ float wlds[];               // [64][KC] zero-padded W strip
  const int tid  = threadIdx.x;
  const int lane = tid & 31;
  const int wib  = tid >> 5;                    // wave in block (0..7)
  const int r    = lane & 15;
  const int half = lane >> 4;

  const int so = (p.O + 63) >> 6;               // 64-wide output strips
  const int s  = blockIdx.x % so;
  const int g  = blockIdx.x / so;
  const int o0 = s << 6;
  const int m0 = (g * 8 + wib) << 4;            // this wave's 16-row tile
  const bool mlive = m0 < p.M;

  // per-lane row context (clamped so all addresses stay in-bounds)
  const int mr   = m0 + r;
  const bool mok = mr < p.M;
  const int mrc  = imin_i(mr, p.M - 1);
  int gn = 0, gj = 0;
  if (MODE == 1) { gn = mrc / p.kk; gj = p.IDX[mrc]; }
  if (MODE == 2) { gn = mrc / p.P;  gj = mrc - gn * p.P; }

  // branchless A fetch: clamped loads + value select (lowers to v_cndmask)
  auto fetchA = [&](int k) -> float {
    const bool kok = k < p.K;
    const int  kc  = imin_i(k, p.K - 1);
    float v;
    if (MODE == 0) {
      v = p.A[(size_t)mrc * p.K + kc];
    } else if (MODE == 1) {
      const bool isDiff = kc < p.C;
      const int  ch     = isDiff ? kc : kc - p.C;
      const float ctr   = p.F[ch * p.N + gn];
      const float nbr   = p.F[ch * p.N + gj];
      v = isDiff ? (nbr - ctr) : ctr;
    } else {
      const bool isDiff = kc < p.C;
      const int  ch     = isDiff ? kc : kc - p.C;
      const float ctr   = p.F[ch * p.P + gn];
      const float nbr   = p.F[ch * p.P + gj];
      v = isDiff ? (nbr - ctr) : ctr;
    }
    return (mok && kok) ? v : 0.f;
  };

  v8f acc0 = {}, acc1 = {}, acc2 = {}, acc3 = {};

  for (int kb = 0; kb < p.K; kb += KC) {
    // cooperative, zero-padded stage of W[o0..o0+63][kb..kb+KC) into LDS
    for (int i = tid; i < 64 * KC; i += 256) {
      const int ol = i >> 7;                    // i / KC
      const int kl = i & (KC - 1);              // i % KC
      const int oc = o0 + ol;
      const int k  = kb + kl;
      const float wv =
          p.W[(size_t)imin_i(oc, p.O - 1) * p.K + imin_i(k, p.K - 1)];
      wlds[ol * KC + kl] = (oc < p.O && k < p.K) ? wv : 0.f;
    }
    __syncthreads();

    if (mlive) {
      const int kcn = (p.K - kb) < KC ? (p.K - kb) : KC;
      for (int k0 = 0; k0 < kcn; k0 += 4) {
        const int ka = k0 + half * 2;           // local K of this half
        v2f av;
        av.x = fetchA(kb + ka);
        av.y = fetchA(kb + ka + 1);
        const float* w0 = wlds + (r)      * KC + ka;
        const float* w1 = wlds + (16 + r) * KC + ka;
        const float* w2 = wlds + (32 + r) * KC + ka;
        const float* w3 = wlds + (48 + r) * KC + ka;
        v2f b0; b0.x = w0[0]; b0.y = w0[1];
        v2f b1; b1.x = w1[0]; b1.y = w1[1];
        v2f b2; b2.x = w2[0]; b2.y = w2[1];
        v2f b3; b3.x = w3[0]; b3.y = w3[1];
        acc0 = __builtin_amdgcn_wmma_f32_16x16x4_f32(false, av, false, b0,
                                                     (short)0, acc0, false, false);
        acc1 = __builtin_amdgcn_wmma_f32_16x16x4_f32(false, av, false, b1,
                                                     (short)0, acc1, false, false);
        acc2 = __builtin_amdgcn_wmma_f32_16x16x4_f32(false, av, false, b2,
                                                     (short)0, acc2, false, false);
        acc3 = __builtin_amdgcn_wmma_f32_16x16x4_f32(false, av, false, b3,
                                                     (short)0, acc3, false, false);
      }
    }
    __syncthreads();
  }

  if (!mlive) return;
  auto epi = [&](const v8f& a, int j) {
    const int oc = o0 + j * 16 + r;
    if (oc >= p.O) return;
    const float sc = p.bg ? p.bg[oc] * BN_INV : 1.f;
    const float bt = p.bb ? p.bb[oc] : 0.f;
#pragma unroll
    for (int i = 0; i < 8; ++i) {
      const int m = m0 + half * 8 + i;
      if (m < p.M) {
        float v = a[i];
        if (p.bg) v = v * sc + bt;
        if (p.act) v = lrelu_f(v);
        p.Out[(size_t)m * p.O + oc] = v;
      }
    }
  };
  epi(acc0, 0); epi(acc1, 1); epi(acc2, 2); epi(acc3, 3);
}

// ---------------------------------------------------------------------------
// Pairwise negative squared distance (one batch), 16x64 strip per wave:
//   D[n,m] = 2*<x_n,x_m> - xx[n] - xx[m],   F = (C,N) feature block
// ---------------------------------------------------------------------------
__global__ void negdist_wmma_k(const float* __restrict__ F,
                               const float* __restrict__ xx,
                               float* __restrict__ D, int N, int C) {
  const int lane   = threadIdx.x & 31;
  const int wave   = (blockIdx.x * blockDim.x + threadIdx.x) >> 5;
  const int nwaves = (gridDim.x * blockDim.x) >> 5;
  const int tm = N >> 4, tc = N >> 6;
  const int r    = lane & 15;
  const int half = lane >> 4;

  for (int t = wave; t < tm * tc; t += nwaves) {
    const int m0 = (t / tc) << 4;
    const int c0 = (t % tc) << 6;
    v8f acc0 = {}, acc1 = {}, acc2 = {}, acc3 = {};
    for (int k0 = 0; k0 < C; k0 += 4) {
      const int ka  = k0 + half * 2;
      const bool k0ok = ka < C, k1ok = ka + 1 < C;
      const int kac  = imin_i(ka, C - 1);
      const int ka1c = imin_i(ka + 1, C - 1);
      v2f av;
      av.x = F[kac * N + m0 + r];  av.x = k0ok ? av.x : 0.f;
      av.y = F[ka1c * N + m0 + r]; av.y = k1ok ? av.y : 0.f;
      v2f b0, b1, b2, b3;
      b0.x = F[kac * N + c0 + r];        b0.x = k0ok ? b0.x : 0.f;
      b0.y = F[ka1c * N + c0 + r];       b0.y = k1ok ? b0.y : 0.f;
      b1.x = F[kac * N + c0 + 16 + r];   b1.x = k0ok ? b1.x : 0.f;
      b1.y = F[ka1c * N + c0 + 16 + r];  b1.y = k1ok ? b1.y : 0.f;
      b2.x = F[kac * N + c0 + 32 + r];   b2.x = k0ok ? b2.x : 0.f;
      b2.y = F[ka1c * N + c0 + 32 + r];  b2.y = k1ok ? b2.y : 0.f;
      b3.x = F[kac * N + c0 + 48 + r];   b3.x = k0ok ? b3.x : 0.f;
      b3.y = F[ka1c * N + c0 + 48 + r];  b3.y = k1ok ? b3.y : 0.f;
      acc0 = __builtin_amdgcn_wmma_f32_16x16x4_f32(false, av, false, b0,
                                                   (short)0, acc0, false, false);
      acc1 = __builtin_amdgcn_wmma_f32_16x16x4_f32(false, av, false, b1,
                                                   (short)0, acc1, false, false);
      acc2 = __builtin_amdgcn_wmma_f32_16x16x4_f32(false, av, false, b2,
                                                   (short)0, acc2, false, false);
      acc3 = __builtin_amdgcn_wmma_f32_16x16x4_f32(false, av, false, b3,
                                                   (short)0, acc3, false, false);
    }
    auto flush = [&](const v8f& a, int j) {
      const int col = c0 + j * 16 + r;
      const float xc = xx[col];
#pragma unroll
      for (int i = 0; i < 8; ++i) {
        const int row = m0 + half * 8 + i;
        D[(size_t)row * N + col] = 2.f * a[i] - xx[row] - xc;
      }
    };
    flush(acc0, 0); flush(acc1, 1); flush(acc2, 2); flush(acc3, 3);
  }
}

// ---------------------------------------------------------------------------
// Small helper kernels
// ---------------------------------------------------------------------------
__global__ void sqnorm_k(const float* __restrict__ f, float* __restrict__ xx,
                         int Bn, int C, int N) {
  int t = blockIdx.x * blockDim.x + threadIdx.x;
  if (t >= Bn * N) return;
  int b = t / N, n = t - b * N;
  float s = 0.f;
  for (int c = 0; c < C; ++c) {
    float v = f[((size_t)b * C + c) * N + n];
    s += v * v;
  }
  xx[t] = s;
}

__global__ void topk_k(const float* __restrict__ D, int* __restrict__ idx, int N) {
  int n = blockIdx.x * blockDim.x + threadIdx.x;
  if (n >= N) return;
  float bv[10];
  int   bi[10];
  for (int i = 0; i < 10; ++i) { bv[i] = -3.4e38f; bi[i] = 0; }
  for (int m = 0; m < N; ++m) {
    float v = D[(size_t)n * N + m];
    if (v > bv[9]) {
      int j = 9;
      while (j > 0 && bv[j - 1] < v) { bv[j] = bv[j - 1]; bi[j] = bi[j - 1]; --j; }
      bv[j] = v; bi[j] = m;
    }
  }
  for (int i = 0; i < 10; ++i) idx[n * 10 + i] = bi[i];
}

// per-edge dynamic matmul + bn + lrelu + max over k  (one batch)
__global__ void edge_max_k(const float* __restrict__ y1,   // [N*kk, O*6]
                           const float* __restrict__ pts,  // (3,N) batch
                           const int*   __restrict__ idx,  // [N,kk]
                           const float* __restrict__ bg, const float* __restrict__ bb,
                           float* __restrict__ xout,       // (O,N) batch slice
                           int N, int kk, int O) {
  int t = blockIdx.x * blockDim.x + threadIdx.x;
  if (t >= N * O) return;
  int o = t / N, n = t - o * N;
  float g = bg[o] * BN_INV, be = bb[o];
  float acc = -3.4e38f;
  float cx = pts[0 * N + n], cy = pts[1 * N + n], cz = pts[2 * N + n];
  for (int k = 0; k < kk; ++k) {
    int j = idx[n * kk + k];
    const float* w = y1 + ((size_t)(n * kk + k)) * (O * 6) + o * 6;
    float s = w[0] * (pts[0 * N + j] - cx) + w[1] * (pts[1 * N + j] - cy) +
              w[2] * (pts[2 * N + j] - cz) + w[3] * cx + w[4] * cy + w[5] * cz;
    s = lrelu_f(s * g + be);
    acc = fmaxf(acc, s);
  }
  xout[(size_t)o * N + n] = acc;
}

__global__ void rowmax_k(const float* __restrict__ h, float* __restrict__ xout,
                         int N, int kk, int O) {
  int t = blockIdx.x * blockDim.x + threadIdx.x;
  if (t >= N * O) return;
  int o = t / N, n = t - o * N;
  float acc = -3.4e38f;
  for (int k = 0; k < kk; ++k)
    acc = fmaxf(acc, h[((size_t)(n * kk + k)) * O + o]);
  xout[(size_t)o * N + n] = acc;
}

__global__ void part_feat_k(const float* __restrict__ f, const int* __restrict__ p2v,
                            const int* __restrict__ pr, float* __restrict__ g,
                            int Bn, int C, int N, int Pn) {
  int t = blockIdx.x * blockDim.x + threadIdx.x;
  if (t >= Bn * C * Pn) return;
  int b = t / (C * Pn);
  int rem = t - b * (C * Pn);
  int c = rem / Pn, p = rem - c * Pn;
  int target = pr[b * Pn + p];
  float s = 0.f; int cnt = 0;
  for (int n = 0; n < N; ++n)
    if (p2v[b * N + n] == target) { s += f[((size_t)b * C + c) * N + n]; ++cnt; }
  g[((size_t)b * C + c) * Pn + p] = s / (float)(cnt < 1 ? 1 : cnt);
}

__global__ void p2p_add_k(float* __restrict__ x, const float* __restrict__ g,
                          const int* __restrict__ p2v, const int* __restrict__ pr,
                          int Bn, int C, int N, int Pn) {
  int t = blockIdx.x * blockDim.x + threadIdx.x;
  if (t >= Bn * C * N) return;
  int b = t / (C * N);
  int rem = t - b * (C * N);
  int c = rem / N, n = rem - c * N;
  int v = p2v[b * N + n];
  int p = 0;
  for (int q = 0; q < Pn; ++q)
    if (pr[b * Pn + q] == v) { p = q; break; }
  x[t] += g[((size_t)b * C + c) * Pn + p];
}

__global__ void pce_k(const float* __restrict__ pc, const float* __restrict__ w,
                      const float* __restrict__ bg, const float* __restrict__ bb,
                      float* __restrict__ out, int Bn, int Pn) {
  int t = blockIdx.x * blockDim.x + threadIdx.x;
  if (t >= Bn * 64 * Pn) return;
  int b = t / (64 * Pn);
  int rem = t - b * (64 * Pn);
  int o = rem / Pn, p = rem - o * Pn;
  float s = 0.f;
  for (int c = 0; c < 3; ++c) s += pc[((size_t)b * 3 + c) * Pn + p] * w[o * 3 + c];
  out[t] = lrelu_f(s * bg[o] * BN_INV + bb[o]);
}

__global__ void add_k(const float* a, const float* b, float* c, int n) {
  int t = blockIdx.x * blockDim.x + threadIdx.x;
  if (t < n) c[t] = a[t] + b[t];
}

// attention over part graph rows: a[h, n, k] = softmax_k( sum_d lrelu(gh*ef) * atw[d] )
__global__ void hop_attn_k(const float* __restrict__ ef, const float* __restrict__ gh,
                           const float* __restrict__ atw, float* __restrict__ a,
                           int Pn, int cout) {
  int t = blockIdx.x * blockDim.x + threadIdx.x;
  if (t >= 4 * Pn) return;
  int h = t / Pn, n = t - h * Pn;
  int d = cout >> 2;
  float s[32];
  float mx = -3.4e38f;
  for (int k = 0; k < Pn; ++k) {
    const float* row = ef + ((size_t)(n * Pn + k)) * cout;
    float gf = gh ? gh[n * Pn + k] : 1.f;
    float acc = 0.f;
    for (int di = 0; di < d; ++di) acc += lrelu_f(gf * row[di * 4 + h]) * atw[di];
    s[k] = acc;
    mx = fmaxf(mx, acc);
  }
  float sum = 0.f;
  for (int k = 0; k < Pn; ++k) { float e = expf(s[k] - mx); s[k] = e; sum += e; }
  float inv = 1.f / sum;
  for (int k = 0; k < Pn; ++k) a[h * Pn * Pn + n * Pn + k] = s[k] * inv;
}

__global__ void attn_apply_k(const float* __restrict__ a, const float* __restrict__ efh,
                             float* __restrict__ ef1, int PP, int cout) {
  int t = blockIdx.x * blockDim.x + threadIdx.x;
  if (t >= PP * cout) return;
  int row = t / cout, c = t - row * cout;
  ef1[t] = a[(c & 3) * PP + row] * efh[t];
}

__global__ void hop_out_k(const float* __restrict__ hl, float* __restrict__ gh,
                          float* __restrict__ hout, int PP) {
  int row = blockIdx.x * blockDim.x + threadIdx.x;
  if (row >= PP) return;
  float best = hl[row * 5];
  int bi = 0;
  for (int c = 1; c < 5; ++c) {
    float v = hl[row * 5 + c];
    if (v > best) { best = v; bi = c; }
  }
  float h = (float)bi;
  gh[row] = expf(-0.5f * h * h) * GAUSS_C;
  for (int c = 0; c < 5; ++c) hout[c * PP + row] = hl[row * 5 + c];
}

__global__ void attn_mean_k(const float* __restrict__ a, const float* __restrict__ ef,
                            float* __restrict__ g, int Pn, int cout) {
  int t = blockIdx.x * blockDim.x + threadIdx.x;
  if (t >= cout * Pn) return;
  int c = t / Pn, n = t - c * Pn;
  float s = 0.f;
  for (int k = 0; k < Pn; ++k)
    s += a[(c & 3) * Pn * Pn + n * Pn + k] * ef[((size_t)(n * Pn + k)) * cout + c];
  g[c * Pn + n] = s / (float)Pn;
}

__global__ void copy_out_k(const float* __restrict__ src, float* __restrict__ dst,
                           int Bn, int C, int N, int Ctot, int coff) {
  int t = blockIdx.x * blockDim.x + threadIdx.x;
  if (t >= Bn * C * N) return;
  int b = t / (C * N);
  int rem = t - b * (C * N);
  int c = rem / N, n = rem - c * N;
  dst[((size_t)b * Ctot + coff + c) * N + n] = src[t];
}

// ---------------------------------------------------------------------------
// Host orchestration
// ---------------------------------------------------------------------------
static void launch_gemm(hipStream_t s, int mode, const GemmP& p) {
  const int tm = (p.M + 15) / 16;
  const int so = (p.O + 63) / 64;
  const int blocks = so * ((tm + 7) / 8);
  const size_t shmem = 64 * KC * sizeof(float);
  if (mode == 0)      gemm_wmma_k<0><<<blocks, 256, shmem, s>>>(p);
  else if (mode == 1) gemm_wmma_k<1><<<blocks, 256, shmem, s>>>(p);
  else                gemm_wmma_k<2><<<blocks, 256, shmem, s>>>(p);
}

struct HGp {
  const float *at1, *at2, *ec1g, *ec1b, *ec1w, *ec2g, *ec2b, *ec2w,
              *h0g, *h0b, *h1g, *h1b, *hw0, *hw1;
};

extern "C" void kernel_launch(void* const* d_in, const int* in_sizes, int n_in,
                              void* d_out, int out_size, void* d_ws, size_t ws_size,
                              hipStream_t stream) {
  (void)in_sizes; (void)n_in; (void)out_size; (void)ws_size;
  const int B = 16, N = 1024, KNN = 10, Pp = 27;

  // ---- input mapping (dict insertion order; params flattened sorted-key) ----
  int ii = 0;
  const float* pts = (const float*)d_in[ii++];                       // x (B,3,N)
  const float* ac1_g0 = (const float*)d_in[ii++]; const float* ac1_b0 = (const float*)d_in[ii++];
  const float* ac1_g1 = (const float*)d_in[ii++]; const float* ac1_b1 = (const float*)d_in[ii++];
  const float* ac1_w0 = (const float*)d_in[ii++]; const float* ac1_w1 = (const float*)d_in[ii++];
  const float* ac2_g0 = (const float*)d_in[ii++]; const float* ac2_b0 = (const float*)d_in[ii++];
  const float* ac2_g1 = (const float*)d_in[ii++]; const float* ac2_b1 = (const float*)d_in[ii++];
  const float* ac2_w0 = (const float*)d_in[ii++]; const float* ac2_w1 = (const float*)d_in[ii++];
  const float* c3_g = (const float*)d_in[ii++]; const float* c3_b = (const float*)d_in[ii++];
  const float* c3_w = (const float*)d_in[ii++];
  const float* c4_g = (const float*)d_in[ii++]; const float* c4_b = (const float*)d_in[ii++];
  const float* c4_w = (const float*)d_in[ii++];
  HGp hg[4];
  for (int q = 0; q < 4; ++q) {
    hg[q].at1  = (const float*)d_in[ii++]; hg[q].at2  = (const float*)d_in[ii++];
    hg[q].ec1g = (const float*)d_in[ii++]; hg[q].ec1b = (const float*)d_in[ii++];
    hg[q].ec1w = (const float*)d_in[ii++];
    hg[q].ec2g = (const float*)d_in[ii++]; hg[q].ec2b = (const float*)d_in[ii++];
    hg[q].ec2w = (const float*)d_in[ii++];
    hg[q].h0g  = (const float*)d_in[ii++]; hg[q].h0b  = (const float*)d_in[ii++];
    hg[q].h1g  = (const float*)d_in[ii++]; hg[q].h1b  = (const float*)d_in[ii++];
    hg[q].hw0  = (const float*)d_in[ii++]; hg[q].hw1  = (const float*)d_in[ii++];
  }
  const float* pce_g = (const float*)d_in[ii++]; const float* pce_b = (const float*)d_in[ii++];
  const float* pce_w = (const float*)d_in[ii++];
  const int* p2v   = (const int*)d_in[ii++];
  const int* prand = (const int*)d_in[ii++];
  float* out = (float*)d_out;

  // ---- workspace carve ----
  char* base = (char*)d_ws;
  size_t off = 0;
  auto ALLOC = [&](size_t nfloats) {
    void* p = base + off;
    off = (off + nfloats * 4 + 255) & ~(size_t)255;
    return p;
  };
  float* xx   = (float*)ALLOC((size_t)B * N);
  float* negd = (float*)ALLOC((size_t)N * N);
  int*   idx  = (int*)  ALLOC((size_t)B * N * KNN);
  float* y0   = (float*)ALLOC((size_t)N * KNN * 64);
  float* y1   = (float*)ALLOC((size_t)N * KNN * 384);   // reused as conv h-buffer
  float* x1b  = (float*)ALLOC((size_t)B * 64 * N);
  float* x2b  = (float*)ALLOC((size_t)B * 64 * N);
  float* x3b  = (float*)ALLOC((size_t)B * 128 * N);
  float* x4b  = (float*)ALLOC((size_t)B * 256 * N);
  float* pcen = (float*)ALLOC((size_t)B * 3 * Pp);
  float* pceo = (float*)ALLOC((size_t)B * 64 * Pp);
  float* pfa  = (float*)ALLOC((size_t)B * 256 * Pp);
  float* pfb  = (float*)ALLOC((size_t)B * 256 * Pp);
  float* gbuf = (float*)ALLOC((size_t)B * 256 * Pp);
  float* ef1h = (float*)ALLOC((size_t)729 * 256);
  float* ef1  = (float*)ALLOC((size_t)729 * 256);
  float* ef2h = (float*)ALLOC((size_t)729 * 256);
  float* h0b  = (float*)ALLOC((size_t)729 * 128);
  float* hlb  = (float*)ALLOC((size_t)729 * 5);
  float* ghb  = (float*)ALLOC((size_t)729);
  float* a1b  = (float*)ALLOC((size_t)4 * 729);

  const size_t OUTSZ = (size_t)B * 512 * N;
  const size_t HOPSZ = (size_t)B * 5 * Pp * Pp;

  auto knn = [&](const float* f, int Cf) {
    sqnorm_k<<<(B * N + 255) / 256, 256, 0, stream>>>(f, xx, B, Cf, N);
    for (int b = 0; b < B; ++b) {
      negdist_wmma_k<<<128, 256, 0, stream>>>(f + (size_t)b * Cf * N, xx + b * N, negd, N, Cf);
      topk_k<<<(N + 255) / 256, 256, 0, stream>>>(negd, idx + (size_t)b * N * KNN, N);
    }
  };

  auto run_ac = [&](const float* f, int Cf,
                    const float* w0, const float* g0, const float* b0,
                    const float* w1, const float* g1, const float* b1,
                    float* xout) {
    knn(f, Cf);
    for (int b = 0; b < B; ++b) {
      GemmP pa = {};
      pa.F = f + (size_t)b * Cf * N; pa.IDX = idx + (size_t)b * N * KNN;
      pa.W = w0; pa.bg = g0; pa.bb = b0; pa.Out = y0;
      pa.M = N * KNN; pa.K = 2 * Cf; pa.O = 64;
      pa.N = N; pa.kk = KNN; pa.C = Cf; pa.P = Pp; pa.act = 1;
      launch_gemm(stream, 1, pa);
      GemmP pb = {};
      pb.A = y0; pb.W = w1; pb.Out = y1;
      pb.M = N * KNN; pb.K = 64; pb.O = 384; pb.act = 0;
      launch_gemm(stream, 0, pb);
      edge_max_k<<<(N * 64 + 255) / 256, 256, 0, stream>>>(
          y1, pts + (size_t)b * 3 * N, idx + (size_t)b * N * KNN,
          g1, b1, xout + (size_t)b * 64 * N, N, KNN, 64);
    }
  };

  auto run_conv = [&](const float* f, int Cf, const float* w, const float* g,
                      const float* bb2, int O, float* xout) {
    knn(f, Cf);
    for (int b = 0; b < B; ++b) {
      GemmP p = {};
      p.F = f + (size_t)b * Cf * N; p.IDX = idx + (size_t)b * N * KNN;
      p.W = w; p.bg = g; p.bb = bb2; p.Out = y1;
      p.M = N * KNN; p.K = 2 * Cf; p.O = O;
      p.N = N; p.kk = KNN; p.C = Cf; p.P = Pp; p.act = 1;
      launch_gemm(stream, 1, p);
      rowmax_k<<<(N * O + 255) / 256, 256, 0, stream>>>(y1, xout + (size_t)b * O * N, N, KNN, O);
    }
  };

  auto run_hop = [&](const float* pf, int C, int cout, const HGp& h,
                     float* gout, float* hout) {
    const int PP = Pp * Pp;
    for (int b = 0; b < B; ++b) {
      const float* pfB = pf + (size_t)b * C * Pp;
      GemmP p1 = {};
      p1.F = pfB; p1.W = h.ec1w; p1.bg = h.ec1g; p1.bb = h.ec1b; p1.Out = ef1h;
      p1.M = PP; p1.K = 2 * C; p1.O = cout; p1.C = C; p1.P = Pp; p1.act = 1;
      launch_gemm(stream, 2, p1);
      hop_attn_k<<<1, 128, 0, stream>>>(ef1h, nullptr, h.at1, a1b, Pp, cout);
      attn_apply_k<<<(PP * cout + 255) / 256, 256, 0, stream>>>(a1b, ef1h, ef1, PP, cout);
      GemmP p2 = {};
      p2.A = ef1; p2.W = h.hw0; p2.bg = h.h0g; p2.bb = h.h0b; p2.Out = h0b;
      p2.M = PP; p2.K = cout; p2.O = cout / 2; p2.act = 1;
      launch_gemm(stream, 0, p2);
      GemmP p3 = {};
      p3.A = h0b; p3.W = h.hw1; p3.bg = h.h1g; p3.bb = h.h1b; p3.Out = hlb;
      p3.M = PP; p3.K = cout / 2; p3.O = 5; p3.act = 1;
      launch_gemm(stream, 0, p3);
      hop_out_k<<<(PP + 255) / 256, 256, 0, stream>>>(hlb, ghb, hout + (size_t)b * 5 * PP, PP);
      GemmP p4 = {};
      p4.A = ef1; p4.W = h.ec2w; p4.bg = h.ec2g; p4.bb = h.ec2b; p4.Out = ef2h;
      p4.M = PP; p4.K = cout; p4.O = cout; p4.act = 1;
      launch_gemm(stream, 0, p4);
      hop_attn_k<<<1, 128, 0, stream>>>(ef2h, ghb, h.at2, a1b, Pp, cout);
      attn_mean_k<<<(cout * Pp + 255) / 256, 256, 0, stream>>>(
          a1b, ef2h, gout + (size_t)b * cout * Pp, Pp, cout);
    }
  };

  auto part_feat = [&](const float* f, int C, float* g) {
    part_feat_k<<<(B * C * Pp + 255) / 256, 256, 0, stream>>>(f, p2v, prand, g, B, C, N, Pp);
  };
  auto p2p_add = [&](float* x, const float* g, int C) {
    p2p_add_k<<<(B * C * N + 255) / 256, 256, 0, stream>>>(x, g, p2v, prand, B, C, N, Pp);
  };

  // ======================= layer 1 =======================
  run_ac(pts, 3, ac1_w0, ac1_g0, ac1_b0, ac1_w1, ac1_g1, ac1_b1, x1b);
  part_feat(pts, 3, pcen);
  pce_k<<<(B * 64 * Pp + 255) / 256, 256, 0, stream>>>(pcen, pce_w, pce_g, pce_b, pceo, B, Pp);
  part_feat(x1b, 64, pfa);
  add_k<<<(B * 64 * Pp + 255) / 256, 256, 0, stream>>>(pceo, pfa, pfb, B * 64 * Pp);
  run_hop(pfb, 64, 64, hg[0], gbuf, out + OUTSZ + 0 * HOPSZ);
  p2p_add(x1b, gbuf, 64);

  // ======================= layer 2 =======================
  run_ac(x1b, 64, ac2_w0, ac2_g0, ac2_b0, ac2_w1, ac2_g1, ac2_b1, x2b);
  part_feat(x2b, 64, pfa);
  run_hop(pfa, 64, 64, hg[1], gbuf, out + OUTSZ + 1 * HOPSZ);
  p2p_add(x2b, gbuf, 64);

  // ======================= layer 3 =======================
  run_conv(x2b, 64, c3_w, c3_g, c3_b, 128, x3b);
  part_feat(x3b, 128, pfa);
  run_hop(pfa, 128, 128, hg[2], gbuf, out + OUTSZ + 2 * HOPSZ);
  p2p_add(x3b, gbuf, 128);

  // ======================= layer 4 =======================
  run_conv(x3b, 128, c4_w, c4_g, c4_b, 256, x4b);
  part_feat(x4b, 256, pfa);
  run_hop(pfa, 256, 256, hg[3], gbuf, out + OUTSZ + 3 * HOPSZ);
  p2p_add(x4b, gbuf, 256);

  // ======================= concat =======================
  copy_out_k<<<(B * 64 * N + 255) / 256, 256, 0, stream>>>(x1b, out, B, 64, N, 512, 0);
  copy_out_k<<<(B * 64 * N + 255) / 256, 256, 0, stream>>>(x2b, out, B, 64, N, 512, 64);
  copy_out_k<<<(B * 128 * N + 255) / 256, 256, 0, stream>>>(x3b, out, B, 128, N, 512, 128);
  copy_out_k<<<(B * 256 * N + 255) / 256, 256, 0, stream>>>(x4b, out, B, 256, N, 512, 256);
}